// StructureEncoder1D_7103875908092
// MI455X (gfx1250) — compile-verified
//
#include <hip/hip_runtime.h>

typedef float v2f __attribute__((ext_vector_type(2)));
typedef float v8f __attribute__((ext_vector_type(8)));

#define NJ 24
#define NF 7
#define FS 6
#define TILE 16
#define WPB 4                           // waves per block (128 threads)
#define WLDS_FLOATS (NJ * 4 * 32 * 2)   // 6144 floats: A-operands (A1lo,A1hi,A2lo,A2hi) x 32 lanes x float2
#define FROW 148                        // dwords per sample row: 144 + 1 float4 pad -> 592 B, 16B-aligned rows
#define FWAVE (TILE * FROW)             // 2368 floats per wave

// single-instruction relu: raw v_max_num_f32 with inline-constant 0 (defeats the
// compiler's maxnum canonicalization pair; inputs are NaN-free so semantics match)
static __device__ __forceinline__ float relu(float v) {
    float r;
    asm("v_max_num_f32 %0, 0, %1" : "=v"(r) : "v"(v));
    return r;
}

// Augmented layer-1 A matrix (16x8): rows 0..6 = W1^T (+bias col at k=7),
// rows 8,9,12 = copies of rows 2,3,6; row 13 = e7 (ones row) so that
// D1 VGPRs {0,1,4,5} are directly the layer-2 B operands (lo=rows0,1,4,5 / hi=rows2,3,6,1).
static __device__ __forceinline__ float a1v(const float* W1, const float* b1, int i, int m, int k) {
    if (m == 13) return (k == 7) ? 1.f : 0.f;
    int ms = (m < 7) ? m : (m == 8 ? 2 : (m == 9 ? 3 : (m == 12 ? 6 : -1)));
    if (ms < 0) return 0.f;
    return (k < 7) ? W1[i * 49 + k * 7 + ms] : b1[i * 7 + ms];
}
// Augmented layer-2 A matrix (16x8): rows 0..5 = W2^T (+bias col), rows 8..13 =
// copies of rows {1,2,5,e7,3,4} so D2 VGPRs give child-layer-1 B halves and
// dual-half feature stores without any cross-lane data movement.
static __device__ __forceinline__ float a2v(const float* W2, const float* b2, int i, int m, int k) {
    if (m == 11) return (k == 7) ? 1.f : 0.f;
    int ms;
    if      (m <  6) ms = m;
    else if (m == 8) ms = 1;
    else if (m == 9) ms = 2;
    else if (m == 10) ms = 5;
    else if (m == 12) ms = 3;
    else if (m == 13) ms = 4;
    else ms = -1;
    if (ms < 0) return 0.f;
    return (k < 7) ? W2[i * 42 + k * 6 + ms] : b2[i * 6 + ms];
}

__global__ __launch_bounds__(128) void se1d_wmma_kernel(
    const float* __restrict__ x, const float* __restrict__ W1,
    const float* __restrict__ b1, const float* __restrict__ W2,
    const float* __restrict__ b2, float* __restrict__ out, int B)
{
    __shared__ float lds[WLDS_FLOATS + WPB * FWAVE];   // 62464 B

    // ---- Stage augmented weights^T into LDS in exact WMMA A-layout.
    // A 16x4 f32: VGPR0 = A[m, half?2:0], VGPR1 = A[m, half?3:1], m = lane&15.
    for (int e = threadIdx.x; e < NJ * 4 * 32; e += blockDim.x) {
        const int i    = e >> 7;
        const int r    = e & 127;
        const int op   = r >> 5;          // 0:A1(K0..3) 1:A1(K4..7) 2:A2(K0..3) 3:A2(K4..7)
        const int lane = r & 31;
        const int m    = lane & 15;
        const int kb   = ((op & 1) ? 4 : 0) + ((lane >> 4) << 1);
        float v0, v1;
        if (op < 2) { v0 = a1v(W1, b1, i, m, kb); v1 = a1v(W1, b1, i, m, kb + 1); }
        else        { v0 = a2v(W2, b2, i, m, kb); v1 = a2v(W2, b2, i, m, kb + 1); }
        lds[2 * e]     = v0;
        lds[2 * e + 1] = v1;
    }
    __syncthreads();

    const int lane = threadIdx.x & 31;
    const int wid  = threadIdx.x >> 5;
    const int tile = blockIdx.x * WPB + wid;
    const int s0   = tile * TILE;
    if (s0 >= B) return;                  // wave-uniform

    const bool lo   = lane < 16;
    const int  sl   = lane & 15;
    const int  srow = s0 + sl;
    const float oneHi = lo ? 0.f : 1.f;

    // per-lane copy of this sample's 24 x-values (both lane halves hold the same sample)
    float xs[NJ];
    {
        const float4* xr = (const float4*)(x + (size_t)srow * NJ);
#pragma unroll
        for (int q = 0; q < 6; ++q) {
            float4 v = xr[q];
            xs[4 * q + 0] = v.x; xs[4 * q + 1] = v.y;
            xs[4 * q + 2] = v.z; xs[4 * q + 3] = v.w;
        }
    }

    float* feat = &lds[WLDS_FLOATS + wid * FWAVE];
    // dual-half feature store pointers: (lo slot / hi slot) per store
    float* fA = feat + sl * FROW + (lo ? 0 : 1);   // stores d2[0]: f0 / f1
    float* fB = feat + sl * FROW + (lo ? 2 : 5);   // stores d2[2]: f2 / f5
    float* fC = feat + sl * FROW + (lo ? 4 : 3);   // stores d2[4]: f4 / f3
    const v2f* wl = (const v2f*)lds;

    static constexpr int kPar[NJ] = {-1,0,0,0,1,2,3,4,5,6,7,8,9,9,9,12,13,14,16,17,18,19,20,21};
    const v8f zacc = {};

    // retained parent accumulator VGPRs (lo half = f0..f4, hi half = {f1,f2,f5,1,f3})
    float K0[NJ], K1[NJ], K2[NJ], K3[NJ], K4[NJ];

#pragma unroll
    for (int i = 0; i < NJ; ++i) {
        const int p = kPar[i];
        v2f a1lo = wl[(i * 4 + 0) * 32 + lane];
        v2f a1hi = wl[(i * 4 + 1) * 32 + lane];
        v2f a2lo = wl[(i * 4 + 2) * 32 + lane];
        v2f a2hi = wl[(i * 4 + 3) * 32 + lane];

        // B rows: 0=x_i, 1..6=parent feats, 7=1.0(bias). Halves come straight
        // from the parent's retained D2 VGPRs (hi halves carry {f1,f2,f5,1}).
        v2f B1a, B1b;
        if (p < 0) {
            B1a.x = lo ? xs[i] : 0.f;
            B1a.y = 0.f;
            B1b.x = 0.f;
            B1b.y = oneHi;
        } else {
            B1a.x = lo ? xs[i] : K0[p];   // x   / pf1
            B1a.y = lo ? K0[p] : K1[p];   // pf0 / pf2
            B1b.x = lo ? K3[p] : K2[p];   // pf3 / pf5
            B1b.y = lo ? K4[p] : K3[p];   // pf4 / 1.0
        }

        v8f d1 = __builtin_amdgcn_wmma_f32_16x16x4_f32(false, a1lo, false, B1a,
                                                       (short)0, zacc, false, false);
        d1 = __builtin_amdgcn_wmma_f32_16x16x4_f32(false, a1hi, false, B1b,
                                                   (short)0, d1, false, false);
        // hidden relu: only the 4 VGPRs that ARE the layer-2 B operands
        v2f B2a = { relu(d1[0]), relu(d1[1]) };   // h0/h2 , h1/h3
        v2f B2b = { relu(d1[4]), relu(d1[5]) };   // h4/h6 , h5/1

        v8f d2 = __builtin_amdgcn_wmma_f32_16x16x4_f32(false, a2lo, false, B2a,
                                                       (short)0, zacc, false, false);
        d2 = __builtin_amdgcn_wmma_f32_16x16x4_f32(false, a2hi, false, B2b,
                                                   (short)0, d2, false, false);

        float f0 = relu(d2[0]);     // f0 / f1
        float f1 = relu(d2[1]);     // f1 / f2
        float f2 = relu(d2[2]);     // f2 / f5
        float f3 = relu(d2[3]);     // f3 / 1
        float f4 = relu(d2[4]);     // f4 / f3
        K0[i] = f0; K1[i] = f1; K2[i] = f2; K3[i] = f3; K4[i] = f4;

        // 3 full-wave stores cover all 6 features (lo: f0,f2,f4 / hi: f1,f5,f3)
        fA[i * FS] = f0;
        fB[i * FS] = f2;
        fC[i * FS] = f4;
    }

    // ---- async LDS -> global writeback: 16 samples x 144 floats, b128 per lane,
    // fully coalesced 512 B per instruction, tracked with ASYNCcnt (no VGPR round trip).
    asm volatile("s_wait_dscnt 0x0" ::: "memory");
    const unsigned long long obase = (unsigned long long)(uintptr_t)(out + (size_t)s0 * (NJ * FS));
    const unsigned lfeat = (unsigned)(uintptr_t)(void*)feat;   // flat->LDS: low 32 bits are the DS address
#pragma unroll
    for (int q0 = 0; q0 < 576; q0 += 32) {
        const int q = q0 + lane;
        const int s = q / 36;                         // sample within tile
        const unsigned voff  = (unsigned)q * 16u;     // global byte offset (contiguous)
        const unsigned laddr = lfeat + voff + (unsigned)s * 16u;  // skip 16B row pad
        asm volatile("global_store_async_from_lds_b128 %0, %1, %2"
                     :: "v"(voff), "v"(laddr), "s"(obase) : "memory");
    }
    asm volatile("s_wait_asynccnt 0x0" ::: "memory");
}

extern "C" void kernel_launch(void* const* d_in, const int* in_sizes, int n_in,
                              void* d_out, int out_size, void* d_ws, size_t ws_size,
                              hipStream_t stream) {
    const float* x  = (const float*)d_in[0];
    const float* W1 = (const float*)d_in[1];
    const float* b1 = (const float*)d_in[2];
    const float* W2 = (const float*)d_in[3];
    const float* b2 = (const float*)d_in[4];
    float* out = (float*)d_out;

    const int B = in_sizes[0] / NJ;
    const int tiles  = (B + TILE - 1) / TILE;
    const int blocks = (tiles + WPB - 1) / WPB;
    se1d_wmma_kernel<<<blocks, 128, 0, stream>>>(x, W1, b1, W2, b2, out, B);
}